// Encoder_2714419331813
// MI455X (gfx1250) — compile-verified
//
#include <hip/hip_runtime.h>
#include <math.h>

#define D 128          // D_IN == D_H == 128

typedef __attribute__((ext_vector_type(2))) float v2f;
typedef __attribute__((ext_vector_type(8))) float v8f;

// ---------------- init / degree / dinv ----------------

__global__ void k_zero(float4* __restrict__ p, long n4) {
    long i = (long)blockIdx.x * blockDim.x + threadIdx.x;
    if (i < n4) p[i] = make_float4(0.f, 0.f, 0.f, 0.f);
}

__global__ void k_init_deg(float* __restrict__ deg, int n) {
    int i = blockIdx.x * blockDim.x + threadIdx.x;
    if (i < n) deg[i] = 1.0f;               // self-loop contributes 1 to every degree
}

__global__ void k_degree(const int* __restrict__ col, float* __restrict__ deg, int E) {
    int e = blockIdx.x * blockDim.x + threadIdx.x;
    if (e < E) atomicAdd(&deg[col[e]], 1.0f);
}

__global__ void k_dinv(const float* __restrict__ deg, float* __restrict__ dinv, int n) {
    int i = blockIdx.x * blockDim.x + threadIdx.x;
    if (i < n) dinv[i] = rsqrtf(deg[i]);    // deg >= 1 always (self-loops)
}

// ---------------- h = x @ W via fp32 WMMA (16x16x4) ----------------
// Per wave: one 16x16 tile of h. Block = 256 threads = 8 waves covering one
// 16-row strip across all 8 column tiles (128 cols).
//
// fp32 A 16x4 layout : lane L: M = L%16, VGPR0 = A[M][k+2*half], VGPR1 = A[M][k+2*half+1]
// fp32 B 4x16 layout : lane L: N = L%16, VGPR0 = B[k+2*half][N], VGPR1 = B[k+2*half+1][N]
// fp32 C/D 16x16     : lane L: N = L%16, VGPR v -> M = v + 8*half

__global__ __launch_bounds__(256) void k_gemm_wmma(const float* __restrict__ x,
                                                   const float* __restrict__ W,
                                                   float* __restrict__ h, int n) {
    const int lane = threadIdx.x & 31;
    const int wave = threadIdx.x >> 5;
    const int m0   = blockIdx.x << 4;
    const int n0   = wave << 4;
    const int lm   = lane & 15;
    const int half = lane >> 4;

    int arow = m0 + lm;
    if (arow >= n) arow = n - 1;            // clamp (N is a multiple of 16 here anyway)
    const float* __restrict__ ap = x + (size_t)arow * D;
    const float* __restrict__ bp = W + n0 + lm;

    v8f acc = {};
#pragma unroll
    for (int k = 0; k < D; k += 4) {
        const int ka = k + 2 * half;
        v2f a, b;
        a[0] = ap[ka];
        a[1] = ap[ka + 1];
        b[0] = bp[(size_t)ka * D];
        b[1] = bp[(size_t)(ka + 1) * D];
        acc = __builtin_amdgcn_wmma_f32_16x16x4_f32(
            /*neg_a=*/false, a, /*neg_b=*/false, b,
            /*c_mod=*/(short)0, acc, /*reuse_a=*/false, /*reuse_b=*/false);
    }

#pragma unroll
    for (int v = 0; v < 8; ++v) {
        const int m = m0 + v + 8 * half;
        if (m < n) h[(size_t)m * D + n0 + lm] = acc[v];
    }
}

// ---------------- edge scatter: out[col] += h[row] * dinv[row]*dinv[col] ----------------
// One wave32 per edge; each lane owns 4 contiguous channels (float4 gather,
// 4 fp32 global atomics). Edge indices are wave-uniform -> scalar loads.

__global__ __launch_bounds__(256) void k_scatter(const int* __restrict__ ei,
                                                 const float* __restrict__ h,
                                                 const float* __restrict__ dinv,
                                                 float* __restrict__ out, int E) {
    const int e = blockIdx.x * (blockDim.x >> 5) + (threadIdx.x >> 5);
    if (e >= E) return;
    const int lane = threadIdx.x & 31;
    const int r = ei[e];          // source node
    const int c = ei[E + e];      // destination node
    const float nrm = dinv[r] * dinv[c];

    const float4 hv = *(const float4*)(h + (size_t)r * D + lane * 4);
    float* __restrict__ o = out + (size_t)c * D + lane * 4;
    atomicAdd(o + 0, hv.x * nrm);
    atomicAdd(o + 1, hv.y * nrm);
    atomicAdd(o + 2, hv.z * nrm);
    atomicAdd(o + 3, hv.w * nrm);
}

// ---------------- finalize: + self-loop + bias, PReLU, row L2 normalize ----------------
// One wave32 per row; lane owns 4 channels; wave32 shuffle reduction for sum of squares.

__global__ __launch_bounds__(256) void k_finalize(float* __restrict__ out,
                                                  const float* __restrict__ h,
                                                  const float* __restrict__ dinv,
                                                  const float* __restrict__ bias,
                                                  const float* __restrict__ alpha,
                                                  int n) {
    const int i = blockIdx.x * (blockDim.x >> 5) + (threadIdx.x >> 5);
    if (i >= n) return;
    const int lane = threadIdx.x & 31;
    const float di = dinv[i];
    const float self = di * di;             // self-loop norm = dinv[i]^2

    float4 o        = *(float4*)(out + (size_t)i * D + lane * 4);
    const float4 hv = *(const float4*)(h + (size_t)i * D + lane * 4);
    const float4 bv = *(const float4*)(bias + lane * 4);
    const float4 av = *(const float4*)(alpha + lane * 4);

    o.x += hv.x * self + bv.x;
    o.y += hv.y * self + bv.y;
    o.z += hv.z * self + bv.z;
    o.w += hv.w * self + bv.w;

    o.x = (o.x >= 0.f) ? o.x : av.x * o.x;
    o.y = (o.y >= 0.f) ? o.y : av.y * o.y;
    o.z = (o.z >= 0.f) ? o.z : av.z * o.z;
    o.w = (o.w >= 0.f) ? o.w : av.w * o.w;

    float ss = o.x * o.x + o.y * o.y + o.z * o.z + o.w * o.w;
#pragma unroll
    for (int off = 16; off > 0; off >>= 1)
        ss += __shfl_xor(ss, off, 32);      // wave32 reduction

    const float sc = 1.0f / fmaxf(sqrtf(ss), 1e-12f);
    o.x *= sc; o.y *= sc; o.z *= sc; o.w *= sc;
    *(float4*)(out + (size_t)i * D + lane * 4) = o;
}

// ---------------- host ----------------

extern "C" void kernel_launch(void* const* d_in, const int* in_sizes, int n_in,
                              void* d_out, int out_size, void* d_ws, size_t ws_size,
                              hipStream_t stream) {
    (void)n_in; (void)out_size; (void)ws_size;
    const float* x     = (const float*)d_in[0];
    const int*   ei    = (const int*)d_in[1];
    const float* W     = (const float*)d_in[2];
    const float* bias  = (const float*)d_in[3];
    const float* alpha = (const float*)d_in[4];
    float* out = (float*)d_out;

    const int N = in_sizes[0] / D;
    const int E = in_sizes[1] / 2;

    // workspace: h [N*D] | deg [N] | dinv [N]
    float* h    = (float*)d_ws;
    float* deg  = h + (size_t)N * D;
    float* dinv = deg + N;

    const int T = 256;
    const long n4 = (long)N * (D / 4);

    k_zero    <<<(int)((n4 + T - 1) / T), T, 0, stream>>>((float4*)out, n4);
    k_init_deg<<<(N + T - 1) / T, T, 0, stream>>>(deg, N);
    k_degree  <<<(E + T - 1) / T, T, 0, stream>>>(ei + E, deg, E);
    k_dinv    <<<(N + T - 1) / T, T, 0, stream>>>(deg, dinv, N);

    k_gemm_wmma<<<(N + 15) / 16, 256, 0, stream>>>(x, W, h, N);

    const int WPB = T / 32;   // waves per block
    k_scatter <<<(E + WPB - 1) / WPB, T, 0, stream>>>(ei, h, dinv, out, E);
    k_finalize<<<(N + WPB - 1) / WPB, T, 0, stream>>>(out, h, dinv, bias, alpha, N);
}